// SimpleMPGNN_85873576116381
// MI455X (gfx1250) — compile-verified
//
#include <hip/hip_runtime.h>
#include <hip/hip_bf16.h>

#define N_NODES   50000
#define N_EDGES   500000
#define NODE_DIM  16
#define EDGE_DIM  8
#define HID       128
#define NUM_GRAPHS 64

typedef __attribute__((ext_vector_type(2))) float v2f;
typedef __attribute__((ext_vector_type(8))) float v8f;

__device__ __forceinline__ v8f wmma_f32(v2f a, v2f b, v8f c) {
  // D = A(16x4) * B(4x16) + C(16x16), exact fp32
  return __builtin_amdgcn_wmma_f32_16x16x4_f32(false, a, false, b, (short)0, c,
                                               false, false);
}

// ---------------------------------------------------------------- zero scratch
__global__ void zero_kernel(float* p, int n) {
  int i = blockIdx.x * blockDim.x + threadIdx.x;
  if (i < n) p[i] = 0.0f;
}

// ------------------------------------------- fold edge encoder into Wm weights
// Wc[8][128]  = We @ Wm3          (Wm3 = rows 256..383 of Wm)
// bc[128]     = be @ Wm3 + bm
__global__ void fold_kernel(const float* __restrict__ We, const float* __restrict__ be,
                            const float* __restrict__ Wm, const float* __restrict__ bm,
                            float* __restrict__ Wc, float* __restrict__ bc) {
  int j = threadIdx.x;  // 0..127 : output column
  if (j >= HID) return;
  for (int i = 0; i < EDGE_DIM; ++i) {
    float s = 0.0f;
    for (int k = 0; k < HID; ++k)
      s += We[i * HID + k] * Wm[(2 * HID + k) * HID + j];
    Wc[i * HID + j] = s;
  }
  float s = bm[j];
  for (int k = 0; k < HID; ++k)
    s += be[k] * Wm[(2 * HID + k) * HID + j];
  bc[j] = s;
}

// --------------------------------------------------------- h = x @ Wn + bn
// grid = 3125 blocks (one 16-node tile), 8 waves = 8 column tiles
__global__ void node_encode_kernel(const float* __restrict__ x,
                                   const float* __restrict__ Wn,
                                   const float* __restrict__ bn,
                                   float* __restrict__ h) {
  const int lane  = threadIdx.x & 31;
  const int wv    = threadIdx.x >> 5;   // column tile 0..7
  const int khalf = lane >> 4;
  const int nlo   = lane & 15;
  const int ncol  = wv * 16 + nlo;
  const int tile  = blockIdx.x;
  const int mg    = tile * 16 + nlo;    // A-matrix row (node)

  float bv = bn[ncol];
  v8f c = {bv, bv, bv, bv, bv, bv, bv, bv};

#pragma unroll
  for (int s = 0; s < 4; ++s) {         // K = 16
    int k = s * 4 + 2 * khalf;
    v2f a = {x[mg * NODE_DIM + k], x[mg * NODE_DIM + k + 1]};
    v2f b = {Wn[k * HID + ncol], Wn[(k + 1) * HID + ncol]};
    c = wmma_f32(a, b, c);
  }
#pragma unroll
  for (int i = 0; i < 8; ++i) {
    int node = tile * 16 + i + 8 * khalf;
    h[node * HID + ncol] = c[i];
  }
}

// ----------------------------------------------- in-degree + per-graph counts
__global__ void counts_kernel(const int* __restrict__ ecol,
                              const int* __restrict__ batch,
                              float* __restrict__ cnt, float* __restrict__ gcnt) {
  int i = blockIdx.x * blockDim.x + threadIdx.x;
  if (i < N_EDGES) atomicAdd(&cnt[ecol[i]], 1.0f);
  if (i < N_NODES) atomicAdd(&gcnt[batch[i]], 1.0f);
}

// ---------------------------------------------------------------- HOT KERNEL
// messages = relu(h[row]@Wm1 + h[col]@Wm2 + edge_attr@Wc + bc)
// scatter-add into agg[col] (L2-resident) — messages never touch memory.
// 256 threads = 8 waves; wave w owns columns 16w..16w+15; B frags live in VGPRs.
#define E_TILES (N_EDGES / 16)
__global__ void edge_msg_kernel(const float* __restrict__ h,
                                const float* __restrict__ ea,
                                const int* __restrict__ erow,
                                const int* __restrict__ ecol,
                                const float* __restrict__ Wm,
                                const float* __restrict__ Wc,
                                const float* __restrict__ bc,
                                float* __restrict__ agg) {
  __shared__ float sh_hr[16 * 132];   // padded stride 132 -> conflict-free
  __shared__ float sh_hc[16 * 132];
  __shared__ float sh_ea[16 * 12];
  __shared__ int   sh_col[16];

  const int lane  = threadIdx.x & 31;
  const int wv    = threadIdx.x >> 5;
  const int khalf = lane >> 4;
  const int nlo   = lane & 15;
  const int ncol  = wv * 16 + nlo;

  // --- preload all B fragments (K=264 -> 66 WMMA steps) into registers ---
  v2f bR[32], bC2[32], bE[2];
#pragma unroll
  for (int s = 0; s < 32; ++s) {
    int k = s * 4 + 2 * khalf;
    bR[s]  = (v2f){Wm[k * HID + ncol],            Wm[(k + 1) * HID + ncol]};
    bC2[s] = (v2f){Wm[(HID + k) * HID + ncol],    Wm[(HID + k + 1) * HID + ncol]};
  }
#pragma unroll
  for (int s = 0; s < 2; ++s) {
    int k = s * 4 + 2 * khalf;
    bE[s] = (v2f){Wc[k * HID + ncol], Wc[(k + 1) * HID + ncol]};
  }
  const float bias = bc[ncol];

  for (int tile = blockIdx.x; tile < E_TILES; tile += gridDim.x) {
    const int e0 = tile * 16;
    __syncthreads();  // protect LDS from previous iteration's readers
    if (threadIdx.x < 16) sh_col[threadIdx.x] = ecol[e0 + threadIdx.x];
    if (threadIdx.x < 128) {
      int r = threadIdx.x >> 3, cc = threadIdx.x & 7;
      sh_ea[r * 12 + cc] = ea[(e0 + r) * EDGE_DIM + cc];
    }
    {
      int r  = threadIdx.x >> 4;        // 0..15 : edge within tile
      int c0 = (threadIdx.x & 15) * 8;  // 8 floats per thread
      int ri = erow[e0 + r];
      int ci = ecol[e0 + r];
#pragma unroll
      for (int j = 0; j < 8; ++j) {
        sh_hr[r * 132 + c0 + j] = h[ri * HID + c0 + j];
        sh_hc[r * 132 + c0 + j] = h[ci * HID + c0 + j];
      }
    }
    __syncthreads();

    v8f c = {bias, bias, bias, bias, bias, bias, bias, bias};
    const int m = nlo;
#pragma unroll
    for (int s = 0; s < 32; ++s) {      // h[row] @ Wm1
      int k = s * 4 + 2 * khalf;
      v2f a = {sh_hr[m * 132 + k], sh_hr[m * 132 + k + 1]};
      c = wmma_f32(a, bR[s], c);
    }
#pragma unroll
    for (int s = 0; s < 32; ++s) {      // h[col] @ Wm2
      int k = s * 4 + 2 * khalf;
      v2f a = {sh_hc[m * 132 + k], sh_hc[m * 132 + k + 1]};
      c = wmma_f32(a, bC2[s], c);
    }
#pragma unroll
    for (int s = 0; s < 2; ++s) {       // edge_attr @ Wc
      int k = s * 4 + 2 * khalf;
      v2f a = {sh_ea[m * 12 + k], sh_ea[m * 12 + k + 1]};
      c = wmma_f32(a, bE[s], c);
    }
    // relu + fused scatter-add into agg (hits L2; agg is 25.6MB)
#pragma unroll
    for (int i = 0; i < 8; ++i) {
      float v = c[i] > 0.0f ? c[i] : 0.0f;
      int rowm = i + 8 * khalf;
      atomicAdd(&agg[sh_col[rowm] * HID + ncol], v);
    }
  }
}

// ------------------- h_upd = relu(h@Wu1 + (agg/cnt)@Wu2 + bu), fused pooling
__global__ void update_pool_kernel(const float* __restrict__ h,
                                   const float* __restrict__ agg,
                                   const float* __restrict__ cnt,
                                   const int* __restrict__ batch,
                                   const float* __restrict__ Wu,
                                   const float* __restrict__ bu,
                                   float* __restrict__ gsum) {
  const int lane  = threadIdx.x & 31;
  const int wv    = threadIdx.x >> 5;
  const int khalf = lane >> 4;
  const int nlo   = lane & 15;
  const int ncol  = wv * 16 + nlo;
  const int tile  = blockIdx.x;
  const int mg    = tile * 16 + nlo;

  float cn  = cnt[mg];
  float inv = 1.0f / (cn > 1.0f ? cn : 1.0f);

  float bv = bu[ncol];
  v8f c = {bv, bv, bv, bv, bv, bv, bv, bv};
#pragma unroll
  for (int s = 0; s < 32; ++s) {        // h @ Wu1
    int k = s * 4 + 2 * khalf;
    v2f a = {h[mg * HID + k], h[mg * HID + k + 1]};
    v2f b = {Wu[k * HID + ncol], Wu[(k + 1) * HID + ncol]};
    c = wmma_f32(a, b, c);
  }
#pragma unroll
  for (int s = 0; s < 32; ++s) {        // (agg/cnt) @ Wu2
    int k = s * 4 + 2 * khalf;
    v2f a = {agg[mg * HID + k] * inv, agg[mg * HID + k + 1] * inv};
    v2f b = {Wu[(HID + k) * HID + ncol], Wu[(HID + k + 1) * HID + ncol]};
    c = wmma_f32(a, b, c);
  }
#pragma unroll
  for (int i = 0; i < 8; ++i) {
    float v = c[i] > 0.0f ? c[i] : 0.0f;
    int node = tile * 16 + i + 8 * khalf;
    atomicAdd(&gsum[batch[node] * HID + ncol], v);  // global mean-pool (sum)
  }
}

// ------------------------------- r1 = relu((gsum/gcnt) @ Wr1 + br1)  [64x128]
__global__ void readout1_kernel(const float* __restrict__ gsum,
                                const float* __restrict__ gcnt,
                                const float* __restrict__ Wr1,
                                const float* __restrict__ br1,
                                float* __restrict__ r1) {
  const int lane  = threadIdx.x & 31;
  const int wv    = threadIdx.x >> 5;
  const int khalf = lane >> 4;
  const int nlo   = lane & 15;
  const int ncol  = wv * 16 + nlo;
  const int tile  = blockIdx.x;      // 0..3
  const int g     = tile * 16 + nlo;

  float cn  = gcnt[g];
  float inv = 1.0f / (cn > 1.0f ? cn : 1.0f);

  float bv = br1[ncol];
  v8f c = {bv, bv, bv, bv, bv, bv, bv, bv};
#pragma unroll
  for (int s = 0; s < 32; ++s) {
    int k = s * 4 + 2 * khalf;
    v2f a = {gsum[g * HID + k] * inv, gsum[g * HID + k + 1] * inv};
    v2f b = {Wr1[k * HID + ncol], Wr1[(k + 1) * HID + ncol]};
    c = wmma_f32(a, b, c);
  }
#pragma unroll
  for (int i = 0; i < 8; ++i) {
    float v = c[i] > 0.0f ? c[i] : 0.0f;
    int gr = tile * 16 + i + 8 * khalf;
    r1[gr * HID + ncol] = v;
  }
}

// ------------------------------------------------- out = r1 @ Wr2 + br2 (OUT=1)
__global__ void readout2_kernel(const float* __restrict__ r1,
                                const float* __restrict__ Wr2,
                                const float* __restrict__ br2,
                                float* __restrict__ out) {
  int g = threadIdx.x;
  if (g >= NUM_GRAPHS) return;
  float s = br2[0];
  for (int k = 0; k < HID; ++k) s += r1[g * HID + k] * Wr2[k];
  out[g] = s;
}

// ============================================================================
extern "C" void kernel_launch(void* const* d_in, const int* in_sizes, int n_in,
                              void* d_out, int out_size, void* d_ws, size_t ws_size,
                              hipStream_t stream) {
  (void)in_sizes; (void)n_in; (void)out_size; (void)ws_size;
  const float* x    = (const float*)d_in[0];
  const float* ea   = (const float*)d_in[1];
  const int*   ei   = (const int*)d_in[2];          // [2, E]
  const int*   batch= (const int*)d_in[3];
  const float* Wn   = (const float*)d_in[4];
  const float* bn   = (const float*)d_in[5];
  const float* We   = (const float*)d_in[6];
  const float* be   = (const float*)d_in[7];
  const float* Wm   = (const float*)d_in[8];
  const float* bm   = (const float*)d_in[9];
  const float* Wu   = (const float*)d_in[10];
  const float* bu   = (const float*)d_in[11];
  const float* Wr1  = (const float*)d_in[12];
  const float* br1  = (const float*)d_in[13];
  const float* Wr2  = (const float*)d_in[14];
  const float* br2  = (const float*)d_in[15];
  float*       out  = (float*)d_out;

  const int* erow = ei;
  const int* ecol = ei + N_EDGES;

  // workspace layout (floats)
  float* ws   = (float*)d_ws;
  float* h    = ws;                       // 50000*128
  float* agg  = h    + N_NODES * HID;     // 50000*128
  float* cnt  = agg  + N_NODES * HID;     // 50000
  float* gsum = cnt  + N_NODES;           // 64*128
  float* gcnt = gsum + NUM_GRAPHS * HID;  // 64
  float* Wc   = gcnt + NUM_GRAPHS;        // 8*128
  float* bc   = Wc   + EDGE_DIM * HID;    // 128
  float* r1   = bc   + HID;               // 64*128

  // 1) zero the accumulators (agg, cnt, gsum, gcnt are contiguous)
  int nz = N_NODES * HID + N_NODES + NUM_GRAPHS * HID + NUM_GRAPHS;
  zero_kernel<<<(nz + 255) / 256, 256, 0, stream>>>(agg, nz);

  // 2) fold edge encoder into message weights
  fold_kernel<<<1, 128, 0, stream>>>(We, be, Wm, bm, Wc, bc);

  // 3) node encoder (WMMA)
  node_encode_kernel<<<N_NODES / 16, 256, 0, stream>>>(x, Wn, bn, h);

  // 4) degree + per-graph counts
  counts_kernel<<<(N_EDGES + 255) / 256, 256, 0, stream>>>(ecol, batch, cnt, gcnt);

  // 5) hot fused edge-message GEMM + scatter (WMMA, B in registers)
  edge_msg_kernel<<<4096, 256, 0, stream>>>(h, ea, erow, ecol, Wm, Wc, bc, agg);

  // 6) node update + graph pooling (WMMA + fused scatter)
  update_pool_kernel<<<N_NODES / 16, 256, 0, stream>>>(h, agg, cnt, batch, Wu, bu, gsum);

  // 7) readout
  readout1_kernel<<<NUM_GRAPHS / 16, 256, 0, stream>>>(gsum, gcnt, Wr1, br1, r1);
  readout2_kernel<<<1, 64, 0, stream>>>(r1, Wr2, br2, out);
}